// GCN_25220047962613
// MI455X (gfx1250) — compile-verified
//
#include <hip/hip_runtime.h>

typedef float v2f __attribute__((ext_vector_type(2)));
typedef float v8f __attribute__((ext_vector_type(8)));

static __device__ __forceinline__ long long gtid() {
    return (long long)blockIdx.x * blockDim.x + threadIdx.x;
}

// ---------------- degree / normalization ----------------

__global__ void k_deg_init(float* deg, int n) {
    int i = blockIdx.x * blockDim.x + threadIdx.x;
    if (i < n) deg[i] = 1.0f;  // self-loop contributes 1 to every node's degree
}

__global__ void k_deg_accum(const int* __restrict__ dst, float* deg, int E) {
    int e = blockIdx.x * blockDim.x + threadIdx.x;
    if (e < E) atomicAdd(&deg[dst[e]], 1.0f);
}

__global__ void k_dinv(float* deg, int n) {
    int i = blockIdx.x * blockDim.x + threadIdx.x;
    if (i < n) {
        float d = deg[i];
        deg[i] = (d > 0.0f) ? rsqrtf(d) : 0.0f;  // in place: deg -> dinv
    }
}

__global__ void k_norm(const int* __restrict__ src, const int* __restrict__ dst,
                       const float* __restrict__ dinv, float* __restrict__ norm, int E) {
    int e = blockIdx.x * blockDim.x + threadIdx.x;
    if (e < E) norm[e] = dinv[src[e]] * dinv[dst[e]];
}

// ---------------- fp32 WMMA GEMM: H[N,OUTW] = A[N,128] @ W[128,OUTW] ----------------
// One 16x16 output tile per wave32. K-loop of 32 steps of V_WMMA_F32_16X16X4_F32.
// Fragment layouts per CDNA5 ISA 7.12.2:
//   A 16x4:  VGPR0 = K0 (lanes 0-15) | K2 (lanes 16-31); VGPR1 = K1 | K3
//   B 4x16:  VGPR0 = row K0 | row K2; VGPR1 = row K1 | row K3 (N = lane%16)
//   C/D:     VGPR r = M=r (lanes 0-15) | M=8+r (lanes 16-31), N = lane%16

template <int OUTW>
__global__ void k_gemm_wmma(const float* __restrict__ A, const float* __restrict__ W,
                            float* __restrict__ H, int totalTiles) {
    const int wave = threadIdx.x >> 5;              // 8 waves per block
    const int lane = threadIdx.x & 31;
    const int tile = blockIdx.x * 8 + wave;
    if (tile >= totalTiles) return;                 // wave-uniform branch: EXEC stays all-1s

    constexpr int NT = OUTW / 16;
    const int m0 = (tile / NT) * 16;
    const int n0 = (tile % NT) * 16;

    const int l15  = lane & 15;
    const int koff = (lane >> 4) * 2;               // 0 for lanes 0-15, 2 for lanes 16-31

    const float* __restrict__ arow = A + (size_t)(m0 + l15) * 128;
    const int col = n0 + l15;

    v8f acc = {};
#pragma unroll 4
    for (int k = 0; k < 128; k += 4) {
        v2f a = *(const v2f*)(arow + k + koff);     // A[row][k+koff], A[row][k+koff+1]
        v2f b;
        b.x = W[(size_t)(k + koff + 0) * OUTW + col];
        b.y = W[(size_t)(k + koff + 1) * OUTW + col];
        acc = __builtin_amdgcn_wmma_f32_16x16x4_f32(
            /*neg_a=*/false, a, /*neg_b=*/false, b,
            /*c_mod=*/(short)0, acc, /*reuse_a=*/false, /*reuse_b=*/false);
    }

    const int srow = m0 + (lane >> 4) * 8;
#pragma unroll
    for (int r = 0; r < 8; ++r)
        H[(size_t)(srow + r) * OUTW + col] = acc[r];
}

// ---------------- self-loop init: agg = dinv^2 * h ----------------

template <int OUTW>
__global__ void k_self_init(const float* __restrict__ dinv, const float* __restrict__ H,
                            float* __restrict__ agg, int N) {
    constexpr int CH = OUTW / 4;
    long long t = gtid();
    if (t >= (long long)N * CH) return;
    int i = (int)(t / CH), c = (int)(t % CH);
    float di = dinv[i];
    float w = di * di;
    float4 v = ((const float4*)(H + (size_t)i * OUTW))[c];
    float4 o = make_float4(w * v.x, w * v.y, w * v.z, w * v.w);
    ((float4*)(agg + (size_t)i * OUTW))[c] = o;
}

// ---------------- edge scatter: agg[dst] += norm * h[src] ----------------

template <int OUTW>
__global__ void k_scatter(const int* __restrict__ src, const int* __restrict__ dst,
                          const float* __restrict__ norm, const float* __restrict__ H,
                          float* agg, int E) {
    constexpr int CH = OUTW / 4;
    long long t = gtid();
    if (t >= (long long)E * CH) return;
    int e = (int)(t / CH), c = (int)(t % CH);
    int s = src[e], d = dst[e];
    float w = norm[e];
    float4 v = ((const float4*)(H + (size_t)s * OUTW))[c];
    float* ap = agg + (size_t)d * OUTW + c * 4;
    atomicAdd(ap + 0, w * v.x);
    atomicAdd(ap + 1, w * v.y);
    atomicAdd(ap + 2, w * v.z);
    atomicAdd(ap + 3, w * v.w);
}

// ---------------- epilogues ----------------

__global__ void k_bias_relu(float* __restrict__ agg, const float* __restrict__ b,
                            long long total, int OUTW) {
    long long t = gtid();
    if (t >= total) return;
    float v = agg[t] + b[(int)(t % OUTW)];
    agg[t] = v > 0.0f ? v : 0.0f;
}

__global__ void k_bias_add(float* __restrict__ out, const float* __restrict__ b,
                           long long total, int OUTW) {
    long long t = gtid();
    if (t >= total) return;
    out[t] = out[t] + b[(int)(t % OUTW)];
}

// ---------------- launch ----------------

extern "C" void kernel_launch(void* const* d_in, const int* in_sizes, int n_in,
                              void* d_out, int out_size, void* d_ws, size_t ws_size,
                              hipStream_t stream) {
    const float* x  = (const float*)d_in[0];
    const int*   ei = (const int*)d_in[1];
    const float* W1 = (const float*)d_in[2];
    const float* b1 = (const float*)d_in[3];
    const float* W2 = (const float*)d_in[4];
    const float* b2 = (const float*)d_in[5];
    float* out = (float*)d_out;

    const int N = in_sizes[0] / 128;   // 100000
    const int E = in_sizes[1] / 2;     // 1600000
    const int* src = ei;
    const int* dst = ei + E;

    // workspace carve-up (256B aligned)
    char* ws = (char*)d_ws;
    size_t o = 0;
    auto alignup = [](size_t v) { return (v + 255) & ~(size_t)255; };
    float* dinv = (float*)(ws + o); o = alignup(o + (size_t)N * 4);
    float* nrm  = (float*)(ws + o); o = alignup(o + (size_t)E * 4);
    float* h1   = (float*)(ws + o); o = alignup(o + (size_t)N * 128 * 4);
    float* agg1 = (float*)(ws + o); o = alignup(o + (size_t)N * 128 * 4);   // becomes x2 in place
    float* h2   = (float*)(ws + o); o = alignup(o + (size_t)N * 64 * 4);
    (void)ws_size; (void)n_in; (void)out_size;

    const int B = 256;
    auto nb = [&](long long total) { return (int)((total + B - 1) / B); };

    // normalization
    k_deg_init<<<nb(N), B, 0, stream>>>(dinv, N);
    k_deg_accum<<<nb(E), B, 0, stream>>>(dst, dinv, E);
    k_dinv<<<nb(N), B, 0, stream>>>(dinv, N);
    k_norm<<<nb(E), B, 0, stream>>>(src, dst, dinv, nrm, E);

    // layer 1: h1 = x @ W1 ; agg1 = scatter(norm * h1[src]) ; x2 = relu(agg1 + b1)
    {
        const int tiles = (N / 16) * (128 / 16);
        k_gemm_wmma<128><<<(tiles + 7) / 8, B, 0, stream>>>(x, W1, h1, tiles);
        k_self_init<128><<<nb((long long)N * 32), B, 0, stream>>>(dinv, h1, agg1, N);
        k_scatter<128><<<nb((long long)E * 32), B, 0, stream>>>(src, dst, nrm, h1, agg1, E);
        k_bias_relu<<<nb((long long)N * 128), B, 0, stream>>>(agg1, b1, (long long)N * 128, 128);
    }

    // layer 2: h2 = x2 @ W2 ; out = scatter(norm * h2[src]) + b2
    {
        const int tiles = (N / 16) * (64 / 16);
        k_gemm_wmma<64><<<(tiles + 7) / 8, B, 0, stream>>>(agg1, W2, h2, tiles);
        k_self_init<64><<<nb((long long)N * 16), B, 0, stream>>>(dinv, h2, out, N);
        k_scatter<64><<<nb((long long)E * 16), B, 0, stream>>>(src, dst, nrm, h2, out, E);
        k_bias_add<<<nb((long long)N * 64), B, 0, stream>>>(out, b2, (long long)N * 64, 64);
    }
}